// RecurrentGCN_GCLSTM_80504866996302
// MI455X (gfx1250) — compile-verified
//
#include <hip/hip_runtime.h>
#include <hip/hip_bf16.h>

typedef __attribute__((ext_vector_type(2))) float v2f;
typedef __attribute__((ext_vector_type(8))) float v8f;
typedef __attribute__((ext_vector_type(2))) int   v2i_;

#define FDIM 256
#define KTOT 1024      // 4 source blocks of 256: [x | h | Tx1 | Tx2]
#define KCH  64        // K chunk staged in LDS (64 divides 256: chunks never cross sources)

#if defined(__HIP_DEVICE_COMPILE__) && __has_builtin(__builtin_amdgcn_global_load_async_to_lds_b64)
#define USE_ASYNC_LDS 1
#else
#define USE_ASYNC_LDS 0
#endif

// ---------------- small helper kernels ----------------

__global__ void fill_zero_kernel(float* __restrict__ p, long n) {
    long i = (long)blockIdx.x * blockDim.x + threadIdx.x;
    if (i < n) p[i] = 0.0f;
}

__global__ void fill_from_kernel(float* __restrict__ p, const float* __restrict__ v, int n) {
    int i = blockIdx.x * blockDim.x + threadIdx.x;
    if (i < n) p[i] = v[0];
}

__global__ void deg_kernel(const int* __restrict__ src, const float* __restrict__ ew,
                           float* __restrict__ deg, int E) {
    int e = blockIdx.x * blockDim.x + threadIdx.x;
    if (e < E) atomicAdd(&deg[src[e]], ew[e]);
}

__global__ void dis_kernel(const float* __restrict__ deg, float* __restrict__ dis, int n) {
    int i = blockIdx.x * blockDim.x + threadIdx.x;
    if (i < n) {
        float d = deg[i];
        dis[i] = (d > 0.0f) ? rsqrtf(fmaxf(d, 1e-12f)) : 0.0f;
    }
}

__global__ void wnorm_kernel(const int* __restrict__ src, const int* __restrict__ dst,
                             const float* __restrict__ ew, const float* __restrict__ dis,
                             float* __restrict__ wn, int E) {
    int e = blockIdx.x * blockDim.x + threadIdx.x;
    if (e < E) wn[e] = -dis[src[e]] * ew[e] * dis[dst[e]];
}

// one edge per block; 256 threads = one feature each (coalesced gather + atomic scatter)
__global__ __launch_bounds__(FDIM)
void scatter_kernel(const int* __restrict__ src, const int* __restrict__ dst,
                    const float* __restrict__ wn, const float* __restrict__ vin,
                    float* __restrict__ vout) {
    int e = blockIdx.x;
    int f = threadIdx.x;
    float w = wn[e];
    int s = src[e], d = dst[e];
    atomicAdd(&vout[(size_t)d * FDIM + f], w * vin[(size_t)s * FDIM + f]);
}

// Tx2 = 2*P - h  (P accumulated in-place in tx2 buffer)
__global__ void tx2_finish_kernel(float* __restrict__ tx2, const float* __restrict__ h, long n) {
    long i = (long)blockIdx.x * blockDim.x + threadIdx.x;
    if (i < n) tx2[i] = 2.0f * tx2[i] - h[i];
}

// Pack big weight matrix G (1024x1024) into K-pair-major layout:
// Gp[(k>>1)*1024 + j] = ( G[k][j], G[k+1][j] )   with j = gate*256 + f
// G rows: k<256 -> W[g][k][f]; else theta[g][blk-1][k%256][f]
__global__ void packB_kernel(const float* __restrict__ W, const float* __restrict__ theta,
                             float2* __restrict__ Gp) {
    long pid = (long)blockIdx.x * blockDim.x + threadIdx.x;   // 0 .. 512*1024-1
    if (pid >= (long)512 * 1024) return;
    int k2 = (int)(pid >> 10);          // K pair index 0..511
    int j  = (int)(pid & 1023);         // output column
    int g  = j >> 8;
    int f  = j & 255;
    int k  = k2 << 1;
    int blk  = k >> 8;
    int krow = k & 255;
    const float* B = (blk == 0) ? (W + (size_t)g * FDIM * FDIM)
                                : (theta + (size_t)(g * 3 + (blk - 1)) * FDIM * FDIM);
    float2 v;
    v.x = B[(size_t)krow * FDIM + f];
    v.y = B[(size_t)(krow + 1) * FDIM + f];
    Gp[pid] = v;
}

__device__ __forceinline__ float sigmoidf_(float x) { return 1.0f / (1.0f + __expf(-x)); }

// ---------------- fused GEMM (WMMA f32) + GCLSTM pointwise + classifier ----------------
// grid.x = N/16 node blocks, grid.y = F/16 feature blocks; block = 128 (4 waves)
// wave g computes gate g's 16x16 tile: Z_g = [x|h|Tx1|Tx2](16x1024) @ [W_g; th_g0; th_g1; th_g2]
__global__ __launch_bounds__(128)
void gclstm_kernel(const float* __restrict__ x,   const float* __restrict__ h,
                   const float* __restrict__ tx1, const float* __restrict__ tx2,
                   const float2* __restrict__ Gp,   // packed weights, (512,1024) pairs
                   const float* __restrict__ bconv, // (4,F)
                   const float* __restrict__ bgate, // (4,F)
                   const float* __restrict__ wc,    // (3,F)
                   const float* __restrict__ c_in,  // (N,F)
                   const float* __restrict__ Wcls,  // (F)
                   float* __restrict__ out,         // (N), pre-filled with b_cls
                   float* __restrict__ h_out, float* __restrict__ c_out,
                   int Nn) {
    const int tid  = threadIdx.x;
    const int gate = tid >> 5;          // wave id == gate id, 0..3
    const int lane = tid & 31;
    const int hi   = lane >> 4;
    const int ln   = lane & 15;
    const int mbase = blockIdx.x * 16;  // node block
    const int fbase = blockIdx.y * 16;  // feature block (within each gate's 256 cols)

    __shared__ float2 As2[(KCH / 2) * 17];   // A chunk as K-pairs, [k2][m], padded stride 17
    __shared__ float  Zs[4 * 16 * 17];       // per-gate 16x16 Z tiles, padded

    v8f acc = {};

    for (int kb = 0; kb < KTOT; kb += KCH) {
        const int blk  = kb >> 8;       // source block 0..3
        const int krow = kb & 255;

        const float* A;
        if      (blk == 0) A = x;
        else if (blk == 1) A = h;
        else if (blk == 2) A = tx1;
        else               A = tx2;

        __syncthreads();                // previous chunk fully consumed
        #pragma unroll
        for (int i = 0; i < 4; ++i) {   // 512 K-pairs / 128 threads
            int pi  = i * 128 + tid;
            int kk2 = pi & 31;          // K pair within chunk
            int m   = pi >> 5;          // node row in tile
            int row = mbase + m; if (row >= Nn) row = Nn - 1;   // clamp (keeps EXEC all-1s)
            const float* gsrc = A + (size_t)row * FDIM + krow + 2 * kk2;
#if USE_ASYNC_LDS
            __builtin_amdgcn_global_load_async_to_lds_b64(
                (__attribute__((address_space(1))) v2i_*)gsrc,
                (__attribute__((address_space(3))) v2i_*)&As2[kk2 * 17 + m], 0, 0);
#else
            As2[kk2 * 17 + m] = *(const float2*)gsrc;
#endif
        }
#if USE_ASYNC_LDS
#if __has_builtin(__builtin_amdgcn_s_wait_asynccnt)
        __builtin_amdgcn_s_wait_asynccnt(0);
#else
        asm volatile("s_wait_asynccnt 0" ::: "memory");
#endif
#endif
        __syncthreads();

        // packed-B base for this chunk: pair row (kb>>1), column gate*256+fbase+ln
        const float2* Brow2 = Gp + (size_t)(kb >> 1) * 1024 + gate * FDIM + fbase + ln;
        #pragma unroll
        for (int k4 = 0; k4 < KCH; k4 += 4) {
            const int kp = (k4 >> 1) + hi;                 // K pair for this lane half
            v2f a = *(const v2f*)&As2[kp * 17 + ln];       // A[m=ln][K=2kp..2kp+1]  (ds_load_b64)
            v2f b = *(const v2f*)&Brow2[(size_t)kp * 1024];// B[K=2kp..2kp+1][n=ln]  (global_load_b64)
            acc = __builtin_amdgcn_wmma_f32_16x16x4_f32(
                false, a, false, b, (short)0, acc, false, false);
        }
    }

    // stage Z tiles: acc[r] holds Z[m = r + 8*hi][n = ln]
    #pragma unroll
    for (int r = 0; r < 8; ++r) {
        int m = r + 8 * hi;
        Zs[gate * (16 * 17) + m * 17 + ln] = acc[r];
    }
    __syncthreads();

    // pointwise LSTM over the 16x16 (node, feature) tile: 2 elements / thread
    float partial[2];
    #pragma unroll
    for (int e = 0; e < 2; ++e) {
        int idx  = tid + e * 128;
        int m    = idx >> 4;
        int n    = idx & 15;
        int f    = fbase + n;
        int node = mbase + m;
        bool ok  = (node < Nn);

        float z0 = Zs[0 * 272 + m * 17 + n] + bconv[0 * FDIM + f] + bgate[0 * FDIM + f];
        float z1 = Zs[1 * 272 + m * 17 + n] + bconv[1 * FDIM + f] + bgate[1 * FDIM + f];
        float z2 = Zs[2 * 272 + m * 17 + n] + bconv[2 * FDIM + f] + bgate[2 * FDIM + f];
        float z3 = Zs[3 * 272 + m * 17 + n] + bconv[3 * FDIM + f] + bgate[3 * FDIM + f];

        float cold = ok ? c_in[(size_t)node * FDIM + f] : 0.0f;
        float ig = sigmoidf_(z0 + wc[0 * FDIM + f] * cold);
        float fg = sigmoidf_(z1 + wc[1 * FDIM + f] * cold);
        float tg = tanhf(z2);
        float cn = fg * cold + ig * tg;
        float og = sigmoidf_(z3 + wc[2 * FDIM + f] * cn);
        float hn = og * tanhf(cn);

        if (ok) {
            c_out[(size_t)node * FDIM + f] = cn;
            h_out[(size_t)node * FDIM + f] = hn;
        }
        partial[e] = ok ? fmaxf(hn, 0.0f) * Wcls[f] : 0.0f;
    }

    // classifier partial sums: reduce over the 16 features of this block per node
    __syncthreads();
    #pragma unroll
    for (int e = 0; e < 2; ++e) {
        int idx = tid + e * 128;
        Zs[(idx >> 4) * 17 + (idx & 15)] = partial[e];
    }
    __syncthreads();
    if (tid < 16) {
        float s = 0.0f;
        #pragma unroll
        for (int n = 0; n < 16; ++n) s += Zs[tid * 17 + n];
        if (mbase + tid < Nn) atomicAdd(&out[mbase + tid], s);
    }
}

// ---------------- launcher ----------------

extern "C" void kernel_launch(void* const* d_in, const int* in_sizes, int n_in,
                              void* d_out, int out_size, void* d_ws, size_t ws_size,
                              hipStream_t stream) {
    const float* x     = (const float*)d_in[0];
    const int*   ei    = (const int*)  d_in[1];   // (2,E)
    const float* ew    = (const float*)d_in[2];
    const float* h     = (const float*)d_in[3];
    const float* c     = (const float*)d_in[4];
    const float* W     = (const float*)d_in[5];
    const float* theta = (const float*)d_in[6];
    const float* bconv = (const float*)d_in[7];
    const float* bgate = (const float*)d_in[8];
    const float* wc    = (const float*)d_in[9];
    const float* Wcls  = (const float*)d_in[10];
    const float* bcls  = (const float*)d_in[11];

    const int N = in_sizes[0] / FDIM;
    const int E = in_sizes[2];
    const int* src = ei;
    const int* dst = ei + E;

    float* out  = (float*)d_out;                 // (N,1)
    float* hout = out + N;                       // (N,F)
    float* cout = hout + (size_t)N * FDIM;       // (N,F)

    float* ws  = (float*)d_ws;
    float* deg = ws;
    float* dis = deg + N;
    float* wn  = dis + N;
    float* tx1 = wn + E;
    float* tx2 = tx1 + (size_t)N * FDIM;
    float2* gp = (float2*)(tx2 + (size_t)N * FDIM);   // 512*1024 float2 = 4MB

    const long nf = (long)N * FDIM;

    // init: deg=0, tx1/tx2=0 (contiguous), out=b_cls
    fill_zero_kernel<<<(int)((N + 255) / 256), 256, 0, stream>>>(deg, N);
    fill_zero_kernel<<<(int)((2 * nf + 255) / 256), 256, 0, stream>>>(tx1, 2 * nf);
    fill_from_kernel<<<(N + 255) / 256, 256, 0, stream>>>(out, bcls, N);

    // pack weights into K-pair-major layout (L2-resident, reused by all node blocks)
    packB_kernel<<<(512 * 1024) / 256, 256, 0, stream>>>(W, theta, gp);

    // graph normalization
    deg_kernel<<<(E + 255) / 256, 256, 0, stream>>>(src, ew, deg, E);
    dis_kernel<<<(N + 255) / 256, 256, 0, stream>>>(deg, dis, N);
    wnorm_kernel<<<(E + 255) / 256, 256, 0, stream>>>(src, dst, ew, dis, wn, E);

    // Chebyshev propagation (gate-independent, done once)
    scatter_kernel<<<E, FDIM, 0, stream>>>(src, dst, wn, h, tx1);    // Tx1 = L_hat h
    scatter_kernel<<<E, FDIM, 0, stream>>>(src, dst, wn, tx1, tx2);  // P   = L_hat Tx1
    tx2_finish_kernel<<<(int)((nf + 255) / 256), 256, 0, stream>>>(tx2, h, nf); // Tx2 = 2P - h

    // fused GEMM (f32 WMMA) + LSTM + classifier
    dim3 grid((N + 15) / 16, FDIM / 16);
    gclstm_kernel<<<grid, 128, 0, stream>>>(x, h, tx1, tx2, gp, bconv, bgate,
                                            wc, c, Wcls, out, hout, cout, N);
}